// E3nnForce_15960098472097
// MI455X (gfx1250) — compile-verified
//
#include <hip/hip_runtime.h>
#include <math.h>

// ---- problem constants -------------------------------------------------
#define B_      16
#define L_      802
#define C_      50
#define LAYERS_ 6
#define NB_     10
#define LM2_    800
#define T_      (B_ * LM2_)   // 12800 triplets
#define N_      (9 * T_)      // 115200 nodes
#define E_      (8 * T_)      // 102400 edges
#define BAR_L   0.1f
#define MAX_R   0.06f
#define SQRT15  3.872983346207417f
#define SQRT5   2.23606797749979f

typedef __attribute__((ext_vector_type(2))) float v2f;
typedef __attribute__((ext_vector_type(8))) float v8f;
union U8 { v8f v; float f[8]; };

__device__ __forceinline__ v8f wmma4(v2f a, v2f b, v8f c) {
  // V_WMMA_F32_16X16X4_F32 : D = A(16x4) * B(4x16) + C(16x16)
  return __builtin_amdgcn_wmma_f32_16x16x4_f32(false, a, false, b, (short)0, c,
                                               false, false);
}

__device__ __forceinline__ float sigmoidf_(float x) {
  return 1.0f / (1.0f + expf(-x));
}

// 2D position of node k (0..8) of triplet (b, li)
__device__ __forceinline__ void node_pos(const float* __restrict__ y, int b,
                                         int li, int k, float* p) {
  int jw = k / 3, pt = k % 3;
  const float* yy = y + ((size_t)(b * L_ + li + jw)) * 3;
  float cx = yy[0], cy = yy[1], a = yy[2];
  float dx = -BAR_L * sinf(a), dy = BAR_L * cosf(a);
  float s = (pt == 0) ? 1.0f : ((pt == 1) ? 0.0f : -1.0f);
  p[0] = cx + s * dx;
  p[1] = cy + s * dy;
}

// ---- kernel A: geometry -> y1 (E x 3), y2 (E x 5), rfeat (E x 12) -------
__global__ __launch_bounds__(256) void geom_kernel(
    const float* __restrict__ y, float* __restrict__ y1o,
    float* __restrict__ y2o, float* __restrict__ rf) {
  int e = blockIdx.x * blockDim.x + threadIdx.x;
  if (e >= E_) return;
  int t = e >> 3, j = e & 7;
  int b = t / LM2_, li = t % LM2_;
  const int stab[8] = {1, 1, 4, 4, 7, 7, 1, 4};
  const int dtab[8] = {0, 2, 3, 5, 6, 8, 4, 7};
  float ps[2], pd[2];
  node_pos(y, b, li, stab[j], ps);
  node_pos(y, b, li, dtab[j], pd);
  float vx = pd[0] - ps[0], vy = pd[1] - ps[1];  // vz == 0
  float d = sqrtf(vx * vx + vy * vy);
  float inv = 1.0f / (d + 1e-12f);
  float ux = vx * inv, uy = vy * inv, uz = 0.0f;
  y1o[e * 3 + 0] = ux;
  y1o[e * 3 + 1] = uy;
  y1o[e * 3 + 2] = uz;
  y2o[e * 5 + 0] = SQRT15 * ux * uy;
  y2o[e * 5 + 1] = SQRT15 * uy * uz;
  y2o[e * 5 + 2] = 0.5f * SQRT5 * (3.0f * uz * uz - 1.0f);
  y2o[e * 5 + 3] = SQRT15 * ux * uz;
  y2o[e * 5 + 4] = 0.5f * SQRT15 * (ux * ux - uy * uy);
  const float width = MAX_R / (NB_ - 1);
#pragma unroll
  for (int i = 0; i < NB_; ++i) {
    float c = (MAX_R * i) / (NB_ - 1);
    float q = (d - c) / width;
    rf[e * 12 + i] = expf(-q * q);
  }
  rf[e * 12 + 10] = (j < 6) ? 0.0f : 1.0f;  // edge_attr one-hot
  rf[e * 12 + 11] = (j < 6) ? 1.0f : 0.0f;
}

// ---- kernel B: h init: h0 = Wemb[type], h1 = h2 = 0 ---------------------
__global__ __launch_bounds__(256) void hinit_kernel(
    const float* __restrict__ Wemb, float* __restrict__ h) {
  int tid = blockIdx.x * blockDim.x + threadIdx.x;  // over N_*C_
  int n = tid / C_, c = tid % C_;
  int v = ((n % 3) == 1) ? 0 : 1;  // tile([1,0,1]) node type
  size_t base = (size_t)tid * 9;
  h[base] = Wemb[v * C_ + c];
#pragma unroll
  for (int i = 1; i < 9; ++i) h[base + i] = 0.0f;
}

// ---- kernel C: edge MLP, w = silu(rfeat @ Wr1) @ Wr2 (WMMA) -------------
// One wave per 16-edge tile.  grid = 800 blocks x 256 threads (8 waves).
__global__ __launch_bounds__(256) void edge_mlp_wmma(
    const float* __restrict__ rf, const float* __restrict__ Wr1,
    const float* __restrict__ Wr2, int layer, float* __restrict__ wout) {
  __shared__ float H[8][16 * 68];  // per-wave 16x64 hidden tile (stride 68)
  const float* W1 = Wr1 + (size_t)layer * 12 * 64;
  const float* W2 = Wr2 + (size_t)layer * 64 * 400;
  __builtin_prefetch(W2, 0, 0);

  int wv = threadIdx.x >> 5;
  int lane = threadIdx.x & 31;
  int lh = lane >> 4;   // 0..1  (K-split half)
  int ll = lane & 15;   // 0..15 (M for A, N for B/C)
  int tileId = blockIdx.x * 8 + wv;
  int e0 = tileId * 16;

  // A operand (rfeat rows) for the 3 K-chunks of GEMM1
  v2f a1[3];
#pragma unroll
  for (int kc = 0; kc < 3; ++kc) {
    int k = kc * 4 + 2 * lh;
    const float* row = rf + (size_t)(e0 + ll) * 12;
    a1[kc].x = row[k];
    a1[kc].y = row[k + 1];
  }

  // GEMM1: (16x12) @ (12x64), silu, stage into LDS in row-major
#pragma unroll
  for (int nt = 0; nt < 4; ++nt) {
    U8 acc;
    acc.v = (v8f){0.f, 0.f, 0.f, 0.f, 0.f, 0.f, 0.f, 0.f};
#pragma unroll
    for (int kc = 0; kc < 3; ++kc) {
      int k = kc * 4 + 2 * lh;
      v2f b;
      b.x = W1[k * 64 + nt * 16 + ll];
      b.y = W1[(k + 1) * 64 + nt * 16 + ll];
      acc.v = wmma4(a1[kc], b, acc.v);
    }
#pragma unroll
    for (int v = 0; v < 8; ++v) {
      int m = v + 8 * lh;
      float x = acc.f[v];
      H[wv][m * 68 + nt * 16 + ll] = x * sigmoidf_(x);  // silu
    }
  }
  __syncthreads();

  // Reload hidden tile as A operand (16 K-chunks of 4)
  float ax[16], ay[16];
#pragma unroll
  for (int kc = 0; kc < 16; ++kc) {
    int k = kc * 4 + 2 * lh;
    ax[kc] = H[wv][ll * 68 + k];
    ay[kc] = H[wv][ll * 68 + k + 1];
  }

  // GEMM2: (16x64) @ (64x400) -> w tile
  for (int nt = 0; nt < 25; ++nt) {
    U8 acc;
    acc.v = (v8f){0.f, 0.f, 0.f, 0.f, 0.f, 0.f, 0.f, 0.f};
#pragma unroll
    for (int kc = 0; kc < 16; ++kc) {
      int k = kc * 4 + 2 * lh;
      v2f a;
      a.x = ax[kc];
      a.y = ay[kc];
      v2f b;
      b.x = W2[k * 400 + nt * 16 + ll];
      b.y = W2[(k + 1) * 400 + nt * 16 + ll];
      acc.v = wmma4(a, b, acc.v);
    }
#pragma unroll
    for (int v = 0; v < 8; ++v) {
      int m = v + 8 * lh;
      wout[(size_t)(e0 + m) * 400 + nt * 16 + ll] = acc.f[v];
    }
  }
}

// ---- kernel D: messages (pure gather; each node has <=1 incoming edge) --
__global__ __launch_bounds__(256) void message_kernel(
    const float* __restrict__ h, const float* __restrict__ w,
    const float* __restrict__ y1, const float* __restrict__ y2,
    float* __restrict__ z) {
  int tid = blockIdx.x * blockDim.x + threadIdx.x;  // over N_*C_
  int n = tid / C_, c = tid % C_;
  int k = n % 9, t = n / 9;
  size_t zb = (size_t)tid * 9;
  if (k == 1) {  // node 1 of a triplet receives no edge
#pragma unroll
    for (int i = 0; i < 9; ++i) z[zb + i] = 0.0f;
    return;
  }
  const int jtab[9] = {0, -1, 1, 2, 6, 3, 4, 7, 5};
  const int srct[9] = {1, -1, 1, 4, 1, 4, 7, 4, 7};
  int e = t * 8 + jtab[k];
  int nsrc = t * 9 + srct[k];

  float w8[8];
#pragma unroll
  for (int i = 0; i < 8; ++i) w8[i] = w[(size_t)e * 400 + i * C_ + c];
  float u1[3], u2[5];
#pragma unroll
  for (int i = 0; i < 3; ++i) u1[i] = y1[e * 3 + i];
#pragma unroll
  for (int i = 0; i < 5; ++i) u2[i] = y2[e * 5 + i];

  size_t sb = ((size_t)nsrc * C_ + c) * 9;
  float s0 = h[sb];
  float s1[3], s2[5];
#pragma unroll
  for (int i = 0; i < 3; ++i) s1[i] = h[sb + 1 + i];
#pragma unroll
  for (int i = 0; i < 5; ++i) s2[i] = h[sb + 4 + i];

  float d1 = s1[0] * u1[0] + s1[1] * u1[1] + s1[2] * u1[2];
  float d2 = s2[0] * u2[0] + s2[1] * u2[1] + s2[2] * u2[2] +
             s2[3] * u2[3] + s2[4] * u2[4];
  float cr[3];
  cr[0] = s1[1] * u1[2] - s1[2] * u1[1];
  cr[1] = s1[2] * u1[0] - s1[0] * u1[2];
  cr[2] = s1[0] * u1[1] - s1[1] * u1[0];

  z[zb] = w8[0] * s0 + w8[1] * d1 + w8[2] * d2;
#pragma unroll
  for (int i = 0; i < 3; ++i)
    z[zb + 1 + i] = w8[3] * s0 * u1[i] + w8[4] * s1[i] + w8[5] * cr[i];
#pragma unroll
  for (int i = 0; i < 5; ++i)
    z[zb + 4 + i] = w8[6] * s0 * u2[i] + w8[7] * s2[i];
}

// ---- kernel E: self-interaction  z += Ws[type] @ h  (WMMA) --------------
// Work unit = (type t, component i, 16-node column tile); 64800 units,
// grid = 8100 blocks x 8 waves.  A = Ws (packed 50x50 in LDS, staged with
// GLOBAL_LOAD_ASYNC_TO_LDS_B128), B = h columns held in regs across the 4
// M-tiles, C initialized from z (fuses messages + self-interaction).
__global__ __launch_bounds__(256) void node_update_wmma(
    const float* __restrict__ Ws0, const float* __restrict__ Ws1,
    const float* __restrict__ Ws2, int layer, const float* __restrict__ h,
    float* __restrict__ z) {
  // [wsel*2 + t][c][d] packed: 6 matrices x 2500 floats = 60000 bytes
  __shared__ __align__(16) float Amat[6 * 2500];

  // Async DMA the three contiguous 20000-byte layer chunks into LDS.
  // LDS byte offset = low 32 bits of the flat shared-memory address.
  unsigned ldsBase = (unsigned)(uintptr_t)(&Amat[0]);
  for (int idx = threadIdx.x; idx < 3750; idx += 256) {  // 3750 x 16B
    int o = idx * 16;
    int r = o / 20000;               // 0 -> Ws0, 1 -> Ws1, 2 -> Ws2
    int ro = o - r * 20000;
    const char* gsrc =
        (const char*)((r == 0) ? Ws0 : ((r == 1) ? Ws1 : Ws2)) +
        (size_t)layer * 20000 + ro;
    unsigned ldsAddr = ldsBase + (unsigned)o;
    asm volatile("global_load_async_to_lds_b128 %0, %1, off"
                 :: "v"(ldsAddr), "v"((unsigned long long)(uintptr_t)gsrc)
                 : "memory");
  }
  asm volatile("s_wait_asynccnt 0x0" ::: "memory");
  __syncthreads();

  int wv = threadIdx.x >> 5;
  int lane = threadIdx.x & 31;
  int lh = lane >> 4, ll = lane & 15;
  int unit = blockIdx.x * 8 + wv;  // 0..64799

  int t, i, ct;
  if (unit < 9 * 2400) {        // type 0: n%3==1  (N_/3 nodes)
    t = 0; i = unit / 2400; ct = unit % 2400;
  } else {                       // type 1: n%3 in {0,2}
    int u = unit - 9 * 2400;
    t = 1; i = u / 4800; ct = u % 4800;
  }
  int col = ct * 16 + ll;
  int node = (t == 0) ? (3 * col + 1) : (3 * (col >> 1) + (col & 1) * 2);
  int wsel = (i == 0) ? 0 : ((i < 4) ? 1 : 2);
  int aoff = (wsel * 2 + t) * 2500;

  // B operand (h columns) in registers, reused across M-tiles
  float bx[13], by[13];
#pragma unroll
  for (int kc = 0; kc < 13; ++kc) {
    int d0 = kc * 4 + 2 * lh;
    bx[kc] = (d0 < 50) ? h[((size_t)node * C_ + d0) * 9 + i] : 0.0f;
    by[kc] = (d0 + 1 < 50) ? h[((size_t)node * C_ + d0 + 1) * 9 + i] : 0.0f;
  }

#pragma unroll
  for (int mt = 0; mt < 4; ++mt) {
    int mBase = mt * 16;
    U8 acc;
#pragma unroll
    for (int v = 0; v < 8; ++v) {
      int m = mBase + v + 8 * lh;
      acc.f[v] = (m < 50) ? z[((size_t)node * C_ + m) * 9 + i] : 0.0f;
    }
#pragma unroll
    for (int kc = 0; kc < 13; ++kc) {
      int ma = mBase + ll;
      int k = kc * 4 + 2 * lh;
      v2f a;
      a.x = (ma < 50 && k < 50) ? Amat[aoff + ma * 50 + k] : 0.0f;
      a.y = (ma < 50 && k + 1 < 50) ? Amat[aoff + ma * 50 + k + 1] : 0.0f;
      v2f b;
      b.x = bx[kc];
      b.y = by[kc];
      acc.v = wmma4(a, b, acc.v);
    }
#pragma unroll
    for (int v = 0; v < 8; ++v) {
      int m = mBase + v + 8 * lh;
      if (m < 50) z[((size_t)node * C_ + m) * 9 + i] = acc.f[v];
    }
  }
}

// ---- kernel E2: gate: h0 = silu(z0), h1/h2 = z1/z2 * sigmoid(z0) --------
__global__ __launch_bounds__(256) void gate_kernel(const float* __restrict__ z,
                                                   float* __restrict__ h) {
  int tid = blockIdx.x * blockDim.x + threadIdx.x;  // over N_*C_
  size_t base = (size_t)tid * 9;
  float z0 = z[base];
  float g = sigmoidf_(z0);
  h[base] = z0 * g;
#pragma unroll
  for (int i = 1; i < 9; ++i) h[base + i] = z[base + i] * g;
}

// ---- kernel F: output projection + force/torque assembly ----------------
__global__ __launch_bounds__(256) void output_kernel(
    const float* __restrict__ y, const float* __restrict__ Wout,
    const float* __restrict__ h, float* __restrict__ res) {
  int tid = blockIdx.x * blockDim.x + threadIdx.x;  // over B_*L_
  if (tid >= B_ * L_) return;
  int b = tid / L_, lp = tid % L_;
  float* r = res + (size_t)tid * 3;
  if (lp == 0 || lp == L_ - 1) {
    r[0] = 0.0f; r[1] = 0.0f; r[2] = 0.0f;
    return;
  }
  int li = lp - 1;
  int t = b * LM2_ + li;
  float ox[3], oy[3];  // projected h1 (x,y) for nodes k = 3,4,5
#pragma unroll
  for (int q = 0; q < 3; ++q) {
    int k = 3 + q;
    int n = t * 9 + k;
    int v = ((k % 3) == 1) ? 0 : 1;
    float sx = 0.0f, sy = 0.0f;
    for (int c = 0; c < C_; ++c) {
      float wc = Wout[v * C_ + c];
      size_t hb = ((size_t)n * C_ + c) * 9;
      sx += wc * h[hb + 1];
      sy += wc * h[hb + 2];
    }
    ox[q] = sx;
    oy[q] = sy;
  }
  r[0] = ox[0] + ox[1] + ox[2];
  r[1] = oy[0] + oy[1] + oy[2];
  float a1 = y[((size_t)(b * L_ + li + 1)) * 3 + 2];
  float dx = -BAR_L * sinf(a1), dy = BAR_L * cosf(a1);
  // c2((dx,dy), out[k=3]) + c2((-dx,-dy), out[k=5])
  r[2] = dx * oy[0] - dy * ox[0] - dx * oy[2] + dy * ox[2];
}

// ---- host side ----------------------------------------------------------
extern "C" void kernel_launch(void* const* d_in, const int* in_sizes, int n_in,
                              void* d_out, int out_size, void* d_ws,
                              size_t ws_size, hipStream_t stream) {
  const float* y    = (const float*)d_in[0];
  const float* Wemb = (const float*)d_in[1];
  const float* Wr1  = (const float*)d_in[2];
  const float* Wr2  = (const float*)d_in[3];
  const float* Ws0  = (const float*)d_in[4];
  const float* Ws1  = (const float*)d_in[5];
  const float* Ws2  = (const float*)d_in[6];
  const float* Wout = (const float*)d_in[7];
  float* res = (float*)d_out;

  float* ws = (float*)d_ws;
  size_t o_y1 = 0;
  size_t o_y2 = o_y1 + (size_t)E_ * 3;
  size_t o_rf = o_y2 + (size_t)E_ * 5;
  size_t o_w  = o_rf + (size_t)E_ * 12;
  size_t o_h  = o_w + (size_t)E_ * 400;
  size_t o_z  = o_h + (size_t)N_ * 450;
  float* wy1 = ws + o_y1;
  float* wy2 = ws + o_y2;
  float* wrf = ws + o_rf;
  float* ww  = ws + o_w;
  float* wh  = ws + o_h;
  float* wz  = ws + o_z;

  geom_kernel<<<E_ / 256, 256, 0, stream>>>(y, wy1, wy2, wrf);
  hinit_kernel<<<(N_ * C_) / 256, 256, 0, stream>>>(Wemb, wh);

  for (int l = 0; l < LAYERS_; ++l) {
    edge_mlp_wmma<<<(E_ / 16) / 8, 256, 0, stream>>>(wrf, Wr1, Wr2, l, ww);
    message_kernel<<<(N_ * C_) / 256, 256, 0, stream>>>(wh, ww, wy1, wy2, wz);
    node_update_wmma<<<64800 / 8, 256, 0, stream>>>(Ws0, Ws1, Ws2, l, wh, wz);
    gate_kernel<<<(N_ * C_) / 256, 256, 0, stream>>>(wz, wh);
  }

  output_kernel<<<(B_ * L_ + 255) / 256, 256, 0, stream>>>(y, Wout, wh, res);
}